// CFTModule_12575664243189
// MI455X (gfx1250) — compile-verified
//
#include <hip/hip_runtime.h>
#include <hip/hip_bf16.h>
#include <math.h>
#include <stdint.h>

typedef __attribute__((ext_vector_type(2))) float v2f;
typedef __attribute__((ext_vector_type(4))) float v4f;
typedef __attribute__((ext_vector_type(8))) float v8f;

#define BATCH 16
#define HW    256
#define CH    64
#define NPIX  (HW * HW)     // 65536 pixels per batch
#define NDIM  16            // tokens
#define LDSTR 68            // padded LDS row stride (floats) -> conflict-free ds_load_b64

// ---------------------------------------------------------------------------
// Kernel 1: 64x64 average pooling.  tokens[b][c][token] = mean of 64x64 block.
// One block per (token, batch); 1024 threads = 16 channel-quads x 64 rows.
// Each thread streams one row of its block with float4 loads: a wave touches
// 2 x 256B fully-contiguous, fully-consumed regions per instruction.
// ---------------------------------------------------------------------------
__global__ void __launch_bounds__(1024)
pool_kernel(const float* __restrict__ x, float* __restrict__ tokens) {
    __shared__ v4f part[64][16];
    const int token = blockIdx.x;          // 0..15
    const int b     = blockIdx.y;          // 0..15
    const int tid   = threadIdx.x;         // 0..1023
    const int cg    = tid & 15;            // channel quad: channels 4*cg..4*cg+3
    const int row   = tid >> 4;            // 0..63 row within the 64x64 block
    const int ib = token >> 2, jb = token & 3;

    const int h = (ib << 6) + row;
    const float* rp = x + (size_t)b * NPIX * CH
                        + ((size_t)h * HW + (jb << 6)) * CH + (cg << 2);
    v4f s = {};
    for (int w = 0; w < 64; ++w) {
        v4f v = *(const v4f*)(rp + (size_t)w * CH);
        s.x += v.x; s.y += v.y; s.z += v.z; s.w += v.w;
    }
    part[row][cg] = s;
    __syncthreads();
    if (tid < 16) {
        v4f t = {};
        for (int r = 0; r < 64; ++r) {
            v4f v = part[r][tid];
            t.x += v.x; t.y += v.y; t.z += v.z; t.w += v.w;
        }
        const size_t o = ((size_t)b * CH + (tid << 2)) * NDIM + token;
        tokens[o + 0 * NDIM] = t.x * (1.0f / 4096.0f);
        tokens[o + 1 * NDIM] = t.y * (1.0f / 4096.0f);
        tokens[o + 2 * NDIM] = t.z * (1.0f / 4096.0f);
        tokens[o + 3 * NDIM] = t.w * (1.0f / 4096.0f);
    }
}

// ---------------------------------------------------------------------------
// Kernel 2: qk projection + dots + softmax  ->  attn[b][t][c]  (16 x 64 x 64).
// Tiny: one block per batch, thread i owns attention row i (=t). Deterministic
// fixed-order reductions.
// ---------------------------------------------------------------------------
__global__ void __launch_bounds__(64)
attn_kernel(const float* __restrict__ tokens, const float* __restrict__ w_qkv,
            float* __restrict__ attn) {
    __shared__ float qk[64][32];
    const int b = blockIdx.x;
    const int i = threadIdx.x;             // 0..63 (channel index)

    float tok[NDIM];
    for (int n = 0; n < NDIM; ++n)
        tok[n] = tokens[((size_t)b * CH + i) * NDIM + n];
    for (int d = 0; d < 32; ++d) {
        float s = 0.0f;
        for (int n = 0; n < NDIM; ++n) s += tok[n] * w_qkv[d * NDIM + n];
        qk[i][d] = s;
    }
    __syncthreads();

    float q[NDIM];
    for (int n = 0; n < NDIM; ++n) q[n] = qk[i][n];

    float dots[64];
    float m = -INFINITY;
    for (int j = 0; j < 64; ++j) {
        float s = 0.0f;
        for (int n = 0; n < NDIM; ++n) s += q[n] * qk[j][16 + n];
        s *= 0.25f;                        // DIM^-0.5 = 1/4
        dots[j] = s;
        m = fmaxf(m, s);
    }
    float sum = 0.0f;
    for (int j = 0; j < 64; ++j) { dots[j] = expf(dots[j] - m); sum += dots[j]; }
    const float inv = 1.0f / sum;
    float* arow = attn + ((size_t)b * 64 + i) * 64;
    for (int j = 0; j < 64; ++j) arow[j] = dots[j] * inv;
}

// ---------------------------------------------------------------------------
// Kernel 3: out[b,p,t] = gelu( sum_c x[b,p,c] * attn[b,t,c] )
// Per-batch GEMM (65536 x 64) * (64 x 64)^T via V_WMMA_F32_16X16X4_F32.
// Block = 256 threads = 8 waves; each wave owns a 16-pixel M-tile and all 64
// output channels (4 N-tiles).  attn tile is brought into padded LDS with
// GLOBAL_LOAD_ASYNC_TO_LDS_B128 (ASYNCcnt-tracked, no VGPR staging).
// Streaming x loads / out stores are non-temporal so attn stays L2-resident.
// ---------------------------------------------------------------------------
__global__ void __launch_bounds__(256)
out_kernel(const float* __restrict__ x, const float* __restrict__ attn,
           float* __restrict__ out) {
    __shared__ float Bs[64 * LDSTR];       // attn[t][c], row-padded
    const int b   = blockIdx.y;
    const int tid = threadIdx.x;

    // --- async fill of the 16KB attn tile: 1024 float4 units, 4 per thread ---
    {
        const uint64_t gbase  = (uint64_t)(attn + (size_t)b * 64 * 64);
        const uint32_t ldsbas = (uint32_t)(uintptr_t)&Bs[0];
#pragma unroll
        for (int i = 0; i < 4; ++i) {
            const int u  = tid + i * 256;          // float4 unit index
            const int t  = u >> 4;                 // attn row
            const int cq = u & 15;                 // float4 within row
            const uint32_t loff = ldsbas + (uint32_t)(t * LDSTR + (cq << 2)) * 4u;
            const uint32_t goff = (uint32_t)u << 4;
            asm volatile("global_load_async_to_lds_b128 %0, %1, %2"
                         :: "v"(loff), "v"(goff), "s"(gbase) : "memory");
        }
        asm volatile("s_wait_asynccnt 0" ::: "memory");
    }
    __syncthreads();

    const int wave = tid >> 5;
    const int lane = tid & 31;
    const int hi   = lane >> 4;            // K-half select (ISA A-matrix 16x4 layout)
    const int mrow = lane & 15;            // M row within tile / N column index
    const int pix0 = blockIdx.x * 128 + wave * 16;

    // A fragments: lane's half of its pixel row, all 16 K-chunks (16 x b64 loads)
    const float* arow = x + ((size_t)b * NPIX + pix0 + mrow) * CH + 2 * hi;
    float a0[16], a1[16];
#pragma unroll
    for (int kk = 0; kk < 16; ++kk) {
        v2f av = __builtin_nontemporal_load((const v2f*)(arow + kk * 4));
        a0[kk] = av.x; a1[kk] = av.y;
    }

    float* orow = out + ((size_t)b * NPIX + pix0) * CH;
#pragma unroll
    for (int nt = 0; nt < 4; ++nt) {
        v8f acc = {};
        const int t = nt * 16 + mrow;      // B fragment: B[k][n] = attn[t=n][c=k]
        const float* brow = &Bs[t * LDSTR + 2 * hi];
#pragma unroll
        for (int kk = 0; kk < 16; ++kk) {
            v2f av; av.x = a0[kk]; av.y = a1[kk];
            v2f bv = *(const v2f*)(brow + kk * 4);
            acc = __builtin_amdgcn_wmma_f32_16x16x4_f32(
                false, av, false, bv, (short)0, acc, false, false);
        }
        // D layout: VGPR r -> pixel row (r + 8*hi), column t = nt*16 + mrow
#pragma unroll
        for (int r = 0; r < 8; ++r) {
            float v = acc[r];
            v = 0.5f * v * (1.0f + erff(v * 0.70710678118654752f));  // exact GELU
            __builtin_nontemporal_store(
                v, &orow[(size_t)(r + 8 * hi) * CH + nt * 16 + mrow]);
        }
    }
}

// ---------------------------------------------------------------------------
extern "C" void kernel_launch(void* const* d_in, const int* in_sizes, int n_in,
                              void* d_out, int out_size, void* d_ws, size_t ws_size,
                              hipStream_t stream) {
    const float* x      = (const float*)d_in[0];   // (16,256,256,64) f32
    const float* w_qkv  = (const float*)d_in[1];   // (32,16) f32
    float* out = (float*)d_out;                    // (16,256,256,64) f32

    float* tokens = (float*)d_ws;                  // 16*64*16 f32  (64 KB)
    float* attn   = tokens + BATCH * CH * NDIM;    // 16*64*64 f32  (256 KB)

    pool_kernel<<<dim3(16, BATCH), 1024, 0, stream>>>(x, tokens);
    attn_kernel<<<BATCH, 64, 0, stream>>>(tokens, w_qkv, attn);
    out_kernel<<<dim3(NPIX / 128, BATCH), 256, 0, stream>>>(x, attn, out);
}